// AveragePoolingClassLoss_69217692942452
// MI455X (gfx1250) — compile-verified
//
#include <hip/hip_runtime.h>

typedef float f4  __attribute__((ext_vector_type(4)));
typedef float v2f __attribute__((ext_vector_type(2)));
typedef float v8f __attribute__((ext_vector_type(8)));

#define BATCH 32
#define NCH   6
#define NC    5
#define HW    (512 * 512)
#define CHUNKS 32
#define TPB    256
#define PIX_PER_CHUNK (HW / CHUNKS)   // 8192 pixels per block

// ---------------------------------------------------------------------------
// Kernel 1: zero the per-(image,class) accumulator workspace (32 x (5 sums + 5 counts))
// ---------------------------------------------------------------------------
__global__ void apcl_zero(float* __restrict__ ws) {
    int i = threadIdx.x;
    if (i < BATCH * 2 * NC) ws[i] = 0.0f;
}

// ---------------------------------------------------------------------------
// Kernel 2: streaming pass. Per pixel: argmax over 5 channels, p = 1/sum(exp(x-m)),
// accumulate (p, 1) into per-thread registers; wave32 shuffle reduce; LDS atomics;
// one set of global atomics per block.
// ---------------------------------------------------------------------------
__global__ __launch_bounds__(TPB) void apcl_main(const float* __restrict__ logits,
                                                 float* __restrict__ ws) {
    const int b     = blockIdx.y;
    const int chunk = blockIdx.x;
    const int tid   = threadIdx.x;

    __shared__ float lds_acc[2 * NC];
    if (tid < 2 * NC) lds_acc[tid] = 0.0f;
    __syncthreads();

    const float* plane = logits + (long long)b * NCH * HW;
    const int p0 = chunk * PIX_PER_CHUNK;

    float s0 = 0.f, s1 = 0.f, s2 = 0.f, s3 = 0.f, s4 = 0.f;
    float c0 = 0.f, c1 = 0.f, c2 = 0.f, c3 = 0.f, c4 = 0.f;

    for (int p = p0 + tid * 4; p < p0 + PIX_PER_CHUNK; p += TPB * 4) {
        // 5 coalesced non-temporal b128 streams (one per foreground channel plane)
        f4 v0 = __builtin_nontemporal_load((const f4*)(plane + 0 * HW + p));
        f4 v1 = __builtin_nontemporal_load((const f4*)(plane + 1 * HW + p));
        f4 v2 = __builtin_nontemporal_load((const f4*)(plane + 2 * HW + p));
        f4 v3 = __builtin_nontemporal_load((const f4*)(plane + 3 * HW + p));
        f4 v4 = __builtin_nontemporal_load((const f4*)(plane + 4 * HW + p));
#pragma unroll
        for (int j = 0; j < 4; ++j) {
            float x0 = v0[j], x1 = v1[j], x2 = v2[j], x3 = v3[j], x4 = v4[j];
            // argmax, ties -> lowest index (strict >)
            float m = x0; int k = 0;
            if (x1 > m) { m = x1; k = 1; }
            if (x2 > m) { m = x2; k = 2; }
            if (x3 > m) { m = x3; k = 3; }
            if (x4 > m) { m = x4; k = 4; }
            float den = __expf(x0 - m) + __expf(x1 - m) + __expf(x2 - m) +
                        __expf(x3 - m) + __expf(x4 - m);
            float pmax = __builtin_amdgcn_rcpf(den);   // softmax value at argmax
            s0 += (k == 0) ? pmax : 0.f;  c0 += (k == 0) ? 1.f : 0.f;
            s1 += (k == 1) ? pmax : 0.f;  c1 += (k == 1) ? 1.f : 0.f;
            s2 += (k == 2) ? pmax : 0.f;  c2 += (k == 2) ? 1.f : 0.f;
            s3 += (k == 3) ? pmax : 0.f;  c3 += (k == 3) ? 1.f : 0.f;
            s4 += (k == 4) ? pmax : 0.f;  c4 += (k == 4) ? 1.f : 0.f;
        }
    }

    // wave32 butterfly reduction of the 10 accumulators
#pragma unroll
    for (int off = 16; off >= 1; off >>= 1) {
        s0 += __shfl_xor(s0, off, 32);  c0 += __shfl_xor(c0, off, 32);
        s1 += __shfl_xor(s1, off, 32);  c1 += __shfl_xor(c1, off, 32);
        s2 += __shfl_xor(s2, off, 32);  c2 += __shfl_xor(c2, off, 32);
        s3 += __shfl_xor(s3, off, 32);  c3 += __shfl_xor(c3, off, 32);
        s4 += __shfl_xor(s4, off, 32);  c4 += __shfl_xor(c4, off, 32);
    }
    if ((tid & 31) == 0) {   // one lane per wave -> LDS float atomics (ds_add_f32)
        atomicAdd(&lds_acc[0], s0);  atomicAdd(&lds_acc[NC + 0], c0);
        atomicAdd(&lds_acc[1], s1);  atomicAdd(&lds_acc[NC + 1], c1);
        atomicAdd(&lds_acc[2], s2);  atomicAdd(&lds_acc[NC + 2], c2);
        atomicAdd(&lds_acc[3], s3);  atomicAdd(&lds_acc[NC + 3], c3);
        atomicAdd(&lds_acc[4], s4);  atomicAdd(&lds_acc[NC + 4], c4);
    }
    __syncthreads();
    if (tid < 2 * NC)
        atomicAdd(&ws[b * 2 * NC + tid], lds_acc[tid]);
}

// ---------------------------------------------------------------------------
// Kernel 3: finalize. One wave (32 lanes), each lane handles 5 of the 160
// (image,class) BCE terms, then a single V_WMMA_F32_16X16X4_F32 with an
// all-ones B matrix performs the cross-lane reduction:
//   D[m][*] = rowsum(A,m) = P_m + P_{m+16}
// Lane 0's 8 D-VGPRs cover rows 0-7, lane 16's cover rows 8-15.
// ---------------------------------------------------------------------------
__global__ __launch_bounds__(32) void apcl_final(const float* __restrict__ ws,
                                                 const float* __restrict__ gt,
                                                 float* __restrict__ out) {
    const int lane = threadIdx.x;   // 0..31, no divergence before WMMA (EXEC all-1s)
    float partial = 0.0f;
#pragma unroll
    for (int i = lane; i < BATCH * NC; i += 32) {
        int b = i / NC, c = i % NC;
        float s = ws[b * 2 * NC + c];
        float n = ws[b * 2 * NC + NC + c];
        float agg = (n > 0.0f) ? (s / n) : 0.0f;       // absent class -> 0
        float lp  = fmaxf(logf(agg),     -100.0f);      // BCELoss log clamp
        float l1  = fmaxf(log1pf(-agg),  -100.0f);
        float g   = gt[i];
        partial += g * lp + (1.0f - g) * l1;
    }

    v2f a;  a[0] = partial; a[1] = 0.0f;   // A 16x4: lane l -> row l%16
    v2f bb; bb[0] = 1.0f;   bb[1] = 1.0f;  // B 4x16: all ones
    v8f acc = {};
    v8f d = __builtin_amdgcn_wmma_f32_16x16x4_f32(
        /*neg_a=*/false, a, /*neg_b=*/false, bb,
        /*c_mod=*/(short)0, acc, /*reuse_a=*/false, /*reuse_b=*/false);

    float rowsum = d[0] + d[1] + d[2] + d[3] + d[4] + d[5] + d[6] + d[7];
    float total  = __shfl(rowsum, 0, 32) + __shfl(rowsum, 16, 32);
    if (lane == 0)
        out[0] = -total / (float)(BATCH * NC);
}

// ---------------------------------------------------------------------------
extern "C" void kernel_launch(void* const* d_in, const int* in_sizes, int n_in,
                              void* d_out, int out_size, void* d_ws, size_t ws_size,
                              hipStream_t stream) {
    (void)in_sizes; (void)n_in; (void)out_size; (void)ws_size;
    const float* logits = (const float*)d_in[0];   // [32, 6, 512, 512] f32
    const float* gt     = (const float*)d_in[1];   // [32, 5] f32
    float* ws  = (float*)d_ws;                     // 320 floats
    float* out = (float*)d_out;                    // 1 float

    apcl_zero<<<dim3(1), dim3(320), 0, stream>>>(ws);
    apcl_main<<<dim3(CHUNKS, BATCH), dim3(TPB), 0, stream>>>(logits, ws);
    apcl_final<<<dim3(1), dim3(32), 0, stream>>>(ws, gt, out);
}